// CannyNet_73108933312839
// MI455X (gfx1250) — compile-verified
//
#include <hip/hip_runtime.h>

typedef float v2f __attribute__((ext_vector_type(2)));
typedef float v8f __attribute__((ext_vector_type(8)));

#define TILE 56          // output tile edge
#define REG  64          // loaded region edge (TILE + 2*4 halo)
#define S    68          // LDS row stride in floats (bank-conflict-free, 16B-aligned rows)
#define IMG  4096

// band value: gw[d] if 0<=d<=4 else 0 (branchless selects)
__device__ __forceinline__ float bsel(int d, float g0, float g1, float g2, float g3, float g4) {
    float v = 0.0f;
    v = (d == 0) ? g0 : v;
    v = (d == 1) ? g1 : v;
    v = (d == 2) ? g2 : v;
    v = (d == 3) ? g3 : v;
    v = (d == 4) ? g4 : v;
    return v;
}

__global__ __launch_bounds__(256)
void canny_fused_kernel(const float* __restrict__ img,
                        const float* __restrict__ gauss,
                        float* __restrict__ out)
{
    __shared__ __align__(16) float sA[REG * S];  // input tile, then blurred (reused)
    __shared__ __align__(16) float sB[REG * S];  // horizontal-blur intermediate
    __shared__ __align__(16) float sM[REG * S];  // channel-accumulated gradient magnitude

    const int tid  = threadIdx.x;
    const int lane = tid & 31;
    const int wave = tid >> 5;
    const int mrow = lane & 15;     // M (or N) index inside 16x16 fragment
    const int hi   = lane >> 4;     // half-wave selector
    const int kb   = hi * 2;        // K base for A/B fragments

    const int tx = blockIdx.x, ty = blockIdx.y;
    const int gx0 = tx * TILE - 4;
    const int gy0 = ty * TILE - 4;
    const bool border = (gx0 < 0) || (gy0 < 0) || (gx0 + REG > IMG) || (gy0 + REG > IMG);

    const float g0 = gauss[0], g1 = gauss[1], g2 = gauss[2], g3 = gauss[3], g4 = gauss[4];

    // Channel-summed sobel responses, kept in registers (56*56/256 = 12.25 -> 13 slots)
    float sgx[13], sgy[13];
#pragma unroll
    for (int j = 0; j < 13; ++j) { sgx[j] = 0.0f; sgy[j] = 0.0f; }

    // Lane-resident constant band fragments: value gw[(4t+kb+p) - mrow].
    // Identical striping serves as A-operand (vertical pass) and B-operand (horizontal pass).
    v2f band[5];
#pragma unroll
    for (int t = 0; t < 5; ++t) {
        int k0 = 4 * t + kb;
        band[t].x = bsel(k0     - mrow, g0, g1, g2, g3, g4);
        band[t].y = bsel(k0 + 1 - mrow, g0, g1, g2, g3, g4);
    }

    for (int ch = 0; ch < 3; ++ch) {
        const float* src = img + (size_t)ch * IMG * IMG;

        // ---- Stage 1: async global -> LDS tile load, zero-padded (ASYNCcnt path) ----
        // 64x64 floats as 1024 float4 transfers; x-boundaries are 4-aligned so a
        // float4 is always fully inside or fully outside the image.
#pragma unroll
        for (int k = 0; k < 4; ++k) {
            int i  = tid + k * 256;          // [0,1024)
            int r  = i >> 4;                 // row 0..63
            int q  = i & 15;                 // float4 column 0..15
            int gr = gy0 + r;
            int gc = gx0 + q * 4;
            unsigned lds_off = (unsigned)(unsigned long long)(&sA[r * S + q * 4]);
            if (gr >= 0 && gr < IMG && gc >= 0 && gc < IMG) {
                unsigned long long ga =
                    (unsigned long long)(src + (size_t)gr * IMG + gc);
                asm volatile("global_load_async_to_lds_b128 %0, %1, off"
                             :: "v"(lds_off), "v"(ga) : "memory");
            } else {
                float4 z = {0.0f, 0.0f, 0.0f, 0.0f};
                *(float4*)&sA[r * S + q * 4] = z;
            }
        }
        asm volatile("s_wait_asynccnt 0" ::: "memory");
        __syncthreads();

        // ---- Stage 2: horizontal Gaussian as banded matmul (WMMA f32 16x16x4) ----
        // HB(16x16) = IN(16x20) x Band(20x16); 16 tiles, 2 per wave.
        for (int tix = wave; tix < 16; tix += 8) {
            int rb = (tix >> 2) * 16;
            int cb = (tix & 3) * 16;
            v8f acc = {};
#pragma unroll
            for (int t = 0; t < 5; ++t) {
                int c0 = cb - 2 + 4 * t + kb;
                int ca = c0 < 0 ? 0 : (c0 > 63 ? 63 : c0);           // clamped fetches only
                int cz = (c0 + 1) < 0 ? 0 : ((c0 + 1) > 63 ? 63 : c0 + 1); // feed unused cols
                v2f a;
                a.x = sA[(rb + mrow) * S + ca];
                a.y = sA[(rb + mrow) * S + cz];
                acc = __builtin_amdgcn_wmma_f32_16x16x4_f32(
                          false, a, false, band[t], (short)0, acc, false, false);
            }
#pragma unroll
            for (int j = 0; j < 8; ++j)
                sB[(rb + j + 8 * hi) * S + cb + mrow] = acc[j];
        }
        __syncthreads();

        // Reference zero-pads the h-blur intermediate before the vertical conv.
        if (border) {
#pragma unroll
            for (int k = 0; k < 16; ++k) {
                int i = tid + k * 256;
                int r = i >> 6, c = i & 63;
                int gr = gy0 + r, gc = gx0 + c;
                if (gr < 0 || gr >= IMG || gc < 0 || gc >= IMG) sB[r * S + c] = 0.0f;
            }
            __syncthreads();
        }

        // ---- Stage 3: vertical Gaussian as banded matmul (band is the A operand) ----
        for (int tix = wave; tix < 16; tix += 8) {
            int rb = (tix >> 2) * 16;
            int cb = (tix & 3) * 16;
            v8f acc = {};
#pragma unroll
            for (int t = 0; t < 5; ++t) {
                int r0 = rb - 2 + 4 * t + kb;
                int ra = r0 < 0 ? 0 : (r0 > 63 ? 63 : r0);
                int rz = (r0 + 1) < 0 ? 0 : ((r0 + 1) > 63 ? 63 : r0 + 1);
                v2f b;
                b.x = sB[ra * S + cb + mrow];
                b.y = sB[rz * S + cb + mrow];
                acc = __builtin_amdgcn_wmma_f32_16x16x4_f32(
                          false, band[t], false, b, (short)0, acc, false, false);
            }
#pragma unroll
            for (int j = 0; j < 8; ++j)
                sA[(rb + j + 8 * hi) * S + cb + mrow] = acc[j];   // blurred overwrites input
        }
        __syncthreads();

        // Reference zero-pads blurred before sobel.
        if (border) {
#pragma unroll
            for (int k = 0; k < 16; ++k) {
                int i = tid + k * 256;
                int r = i >> 6, c = i & 63;
                int gr = gy0 + r, gc = gx0 + c;
                if (gr < 0 || gr >= IMG || gc < 0 || gc >= IMG) sA[r * S + c] = 0.0f;
            }
            __syncthreads();
        }

        // ---- Stage 4: sobel + per-channel L2 magnitude on the 58x58 NMS region ----
        for (unsigned i = tid; i < 58u * 58u; i += 256u) {
            int r = (int)(i / 58u), c = (int)(i % 58u);
            int ly = 3 + r, lx = 3 + c;
            int gr = gy0 + ly, gc = gx0 + lx;
            float m = 0.0f;   // grad_mag is zero outside the image (conv padding)
            if (gr >= 0 && gr < IMG && gc >= 0 && gc < IMG) {
                const float* b = &sA[ly * S + lx];
                float tl = b[-S - 1], tc = b[-S], tr = b[-S + 1];
                float ml = b[-1],                  mr = b[1];
                float bl = b[S - 1],  bc = b[S],   br = b[S + 1];
                float gxv = (tl - tr) + 2.0f * (ml - mr) + (bl - br);
                float gyv = (tl + 2.0f * tc + tr) - (bl + 2.0f * bc + br);
                m = sqrtf(gxv * gxv + gyv * gyv);
            }
            if (ch == 0) sM[ly * S + lx] = m;
            else         sM[ly * S + lx] += m;
        }

        // ---- Stage 5: channel-summed gx/gy at output pixels (registers) ----
#pragma unroll
        for (int j = 0; j < 13; ++j) {
            int p = tid + j * 256;
            if (p < TILE * TILE) {
                int r = (int)((unsigned)p / 56u), c = (int)((unsigned)p % 56u);
                int ly = 4 + r, lx = 4 + c;
                const float* b = &sA[ly * S + lx];
                float tl = b[-S - 1], tc = b[-S], tr = b[-S + 1];
                float ml = b[-1],                  mr = b[1];
                float bl = b[S - 1],  bc = b[S],   br = b[S + 1];
                sgx[j] += (tl - tr) + 2.0f * (ml - mr) + (bl - br);
                sgy[j] += (tl + 2.0f * tc + tr) - (bl + 2.0f * bc + br);
            }
        }
        __syncthreads();   // protects sA/sM before next channel overwrites / final reads
    }

    // ---- Stage 6: orientation, non-max suppression, threshold, binary output ----
    // direction k -> neighbor offset (dy,dx), packed 2 bits each (+1 biased)
    const unsigned dyP = 0x01A9u;  // dy: {0,1,1,1,0,-1,-1,-1}
    const unsigned dxP = 0x901Au;  // dx: {1,1,0,-1,-1,-1,0,1}
    const float kdeg = (float)(180.0 / 3.14159);
#pragma unroll
    for (int j = 0; j < 13; ++j) {
        int p = tid + j * 256;
        if (p >= TILE * TILE) continue;
        int r = (int)((unsigned)p / 56u), c = (int)((unsigned)p % 56u);
        int gr = ty * TILE + r, gc = tx * TILE + c;
        if (gr >= IMG || gc >= IMG) continue;
        int ly = 4 + r, lx = 4 + c;
        float m = sM[ly * S + lx];
        float od = atan2f(sgy[j], sgx[j]) * kdeg;
        int q = (int)rintf((od + 180.0f) / 45.0f);   // round-half-even, q in [0,8]
        int ip = q & 7;
        int in_ = (q + 4) & 7;
        int dy1 = (int)((dyP >> (2 * ip)) & 3u) - 1;
        int dx1 = (int)((dxP >> (2 * ip)) & 3u) - 1;
        int dy2 = (int)((dyP >> (2 * in_)) & 3u) - 1;
        int dx2 = (int)((dxP >> (2 * in_)) & 3u) - 1;
        float pos = m - sM[(ly + dy1) * S + (lx + dx1)];
        float neg = m - sM[(ly + dy2) * S + (lx + dx2)];
        bool keep = (fminf(pos, neg) > 0.0f) && (m >= 10.0f);
        out[(size_t)gr * IMG + gc] = keep ? 1.0f : 0.0f;
    }
}

extern "C" void kernel_launch(void* const* d_in, const int* in_sizes, int n_in,
                              void* d_out, int out_size, void* d_ws, size_t ws_size,
                              hipStream_t stream) {
    (void)in_sizes; (void)n_in; (void)out_size; (void)d_ws; (void)ws_size;
    const float* img   = (const float*)d_in[0];
    const float* gauss = (const float*)d_in[1];   // gauss_h (gauss_v identical taps)
    float* out = (float*)d_out;
    dim3 grid((IMG + TILE - 1) / TILE, (IMG + TILE - 1) / TILE);  // 74 x 74
    canny_fused_kernel<<<grid, dim3(256), 0, stream>>>(img, gauss, out);
}